// GCNEncoder_61718680044347
// MI455X (gfx1250) — compile-verified
//
#include <hip/hip_runtime.h>

// GCN 2-layer forward for MI455X (gfx1250, wave32).
// Dense GEMMs use V_WMMA_F32_16X16X4_F32 (full fp32 matrix pipe).
// Edge gather/scatter uses coalesced float4 loads + L2-resident f32 atomics.

typedef float v2f __attribute__((ext_vector_type(2)));
typedef float v8f __attribute__((ext_vector_type(8)));

#define IN_C  128
#define HID   32
#define OUT_C 16

// ---------------- utility kernels ----------------

__global__ void zero_f32(float* __restrict__ p, long long n) {
  long long i = (long long)blockIdx.x * blockDim.x + threadIdx.x;
  if (i < n) p[i] = 0.0f;
}

__global__ void init_out_bias(float* __restrict__ out, const float* __restrict__ b2,
                              long long n) {
  long long i = (long long)blockIdx.x * blockDim.x + threadIdx.x;
  if (i < n) out[i] = b2[i & (OUT_C - 1)];
}

__global__ void degree_count(const int* __restrict__ col, float* __restrict__ deg, int E) {
  int e = blockIdx.x * blockDim.x + threadIdx.x;
  if (e < E) atomicAdd(&deg[col[e]], 1.0f);
}

__global__ void deg_to_dinv(float* __restrict__ deg, int N) {
  int i = blockIdx.x * blockDim.x + threadIdx.x;
  if (i < N) {
    float d = deg[i];
    deg[i] = (d > 0.0f) ? rsqrtf(d) : 0.0f;
  }
}

// ---------------- GEMM 1: h1 = x @ W1  (N x 128) @ (128 x 32), fp32 WMMA ----------------
// One wave owns a 16-row tile of h1 and both 16-col N-tiles (reuses A loads).
// A-layout (32-bit 16x4): lanes 0-15 K={0,1}, lanes 16-31 K={2,3}, one K per VGPR.
// B-layout (4x16): lane = N; VGPR idx + lane-half = K. D: VGPR v -> M=v (lo half) / v+8 (hi half).

__global__ void gemm1_wmma(const float* __restrict__ x, const float* __restrict__ W1,
                           float* __restrict__ h1, int mtiles, long long N) {
  int wave = (int)(((long long)blockIdx.x * blockDim.x + threadIdx.x) >> 5);
  int lane = threadIdx.x & 31;
  if (wave >= mtiles) return;              // uniform per wave: EXEC stays all-1s

  long long m0 = (long long)wave * 16;
  int lhalf = lane >> 4;                   // 0 or 1
  int l15   = lane & 15;
  long long mrow = m0 + l15;
  if (mrow >= N) mrow = N - 1;             // tail clamp (keeps EXEC full)
  int kh = lhalf * 2;

  v8f c0 = {0.f, 0.f, 0.f, 0.f, 0.f, 0.f, 0.f, 0.f};
  v8f c1 = c0;

  const float* xr = x + mrow * IN_C;
  #pragma unroll 4
  for (int k = 0; k < IN_C; k += 4) {
    v2f a;
    a.x = xr[k + kh];
    a.y = xr[k + kh + 1];
    const float* w0 = W1 + (long long)(k + kh) * HID;
    v2f b0, b1;
    b0.x = w0[l15];
    b0.y = w0[HID + l15];
    b1.x = w0[16 + l15];
    b1.y = w0[HID + 16 + l15];
    c0 = __builtin_amdgcn_wmma_f32_16x16x4_f32(false, a, false, b0, (short)0, c0, false, false);
    c1 = __builtin_amdgcn_wmma_f32_16x16x4_f32(false, a, false, b1, (short)0, c1, false, false);
  }

  long long rbase = m0 + (long long)lhalf * 8;
  float* hp = h1 + rbase * HID + l15;
  if (m0 + 16 <= N) {
    // wave-uniform fast path: straight-line stores, no EXEC manipulation
    #pragma unroll
    for (int v = 0; v < 8; ++v) {
      hp[(long long)v * HID]      = c0[v];
      hp[(long long)v * HID + 16] = c1[v];
    }
  } else {
    #pragma unroll
    for (int v = 0; v < 8; ++v) {
      if (rbase + v < N) {
        hp[(long long)v * HID]      = c0[v];
        hp[(long long)v * HID + 16] = c1[v];
      }
    }
  }
}

// ---------------- GEMM 2: h2 = relu(agg1 + b1) @ W2  (N x 32) @ (32 x 16) ----------------
// Bias + ReLU fused into the A-operand load (saves a full pass over agg1).

__global__ void gemm2_wmma(const float* __restrict__ agg1, const float* __restrict__ b1,
                           const float* __restrict__ W2, float* __restrict__ h2,
                           int mtiles, long long N) {
  int wave = (int)(((long long)blockIdx.x * blockDim.x + threadIdx.x) >> 5);
  int lane = threadIdx.x & 31;
  if (wave >= mtiles) return;

  long long m0 = (long long)wave * 16;
  int lhalf = lane >> 4;
  int l15   = lane & 15;
  long long mrow = m0 + l15;
  if (mrow >= N) mrow = N - 1;
  int kh = lhalf * 2;

  v8f c = {0.f, 0.f, 0.f, 0.f, 0.f, 0.f, 0.f, 0.f};

  const float* ar = agg1 + mrow * HID;
  #pragma unroll
  for (int k = 0; k < HID; k += 4) {
    float a0 = ar[k + kh]     + b1[k + kh];
    float a1 = ar[k + kh + 1] + b1[k + kh + 1];
    v2f a;
    a.x = (a0 > 0.0f) ? a0 : 0.0f;
    a.y = (a1 > 0.0f) ? a1 : 0.0f;
    const float* w0 = W2 + (long long)(k + kh) * OUT_C;
    v2f b;
    b.x = w0[l15];
    b.y = w0[OUT_C + l15];
    c = __builtin_amdgcn_wmma_f32_16x16x4_f32(false, a, false, b, (short)0, c, false, false);
  }

  long long rbase = m0 + (long long)lhalf * 8;
  float* hp = h2 + rbase * OUT_C + l15;
  if (m0 + 16 <= N) {
    #pragma unroll
    for (int v = 0; v < 8; ++v) hp[(long long)v * OUT_C] = c[v];
  } else {
    #pragma unroll
    for (int v = 0; v < 8; ++v)
      if (rbase + v < N) hp[(long long)v * OUT_C] = c[v];
  }
}

// ---------------- edge scatters: agg[col] += h[row] * dinv[row]*dinv[col] ----------------

__global__ void scatter_h1(const int* __restrict__ row, const int* __restrict__ col,
                           const float* __restrict__ dinv, const float* __restrict__ h1,
                           float* __restrict__ agg1, int E) {
  long long t = (long long)blockIdx.x * blockDim.x + threadIdx.x;
  int e = (int)(t >> 3);                      // HID/4 = 8 groups per edge
  if (e >= E) return;
  int f = (int)(t & 7) * 4;
  int r = row[e], c = col[e];
  float nrm = dinv[r] * dinv[c];
  float4 hv = *(const float4*)(h1 + (long long)r * HID + f);
  float* dst = agg1 + (long long)c * HID + f;
  atomicAdd(dst + 0, hv.x * nrm);
  atomicAdd(dst + 1, hv.y * nrm);
  atomicAdd(dst + 2, hv.z * nrm);
  atomicAdd(dst + 3, hv.w * nrm);
}

__global__ void scatter_h2(const int* __restrict__ row, const int* __restrict__ col,
                           const float* __restrict__ dinv, const float* __restrict__ h2,
                           float* __restrict__ out, int E) {
  long long t = (long long)blockIdx.x * blockDim.x + threadIdx.x;
  int e = (int)(t >> 2);                      // OUT_C/4 = 4 groups per edge
  if (e >= E) return;
  int f = (int)(t & 3) * 4;
  int r = row[e], c = col[e];
  float nrm = dinv[r] * dinv[c];
  float4 hv = *(const float4*)(h2 + (long long)r * OUT_C + f);
  float* dst = out + (long long)c * OUT_C + f;
  atomicAdd(dst + 0, hv.x * nrm);
  atomicAdd(dst + 1, hv.y * nrm);
  atomicAdd(dst + 2, hv.z * nrm);
  atomicAdd(dst + 3, hv.w * nrm);
}

// ---------------- launch ----------------

extern "C" void kernel_launch(void* const* d_in, const int* in_sizes, int n_in,
                              void* d_out, int out_size, void* d_ws, size_t ws_size,
                              hipStream_t stream) {
  const float* x  = (const float*)d_in[0];
  const int*   ei = (const int*)d_in[1];
  const float* W1 = (const float*)d_in[2];
  const float* b1 = (const float*)d_in[3];
  const float* W2 = (const float*)d_in[4];
  const float* b2 = (const float*)d_in[5];
  float* out = (float*)d_out;

  long long N = in_sizes[0] / IN_C;     // 100000
  int E = in_sizes[1] / 2;              // 3200000
  const int* row = ei;
  const int* col = ei + E;

  // workspace layout (fp32): deg/dinv | h1 | agg1 | h2  (~32.4 MB total)
  float* ws = (float*)d_ws;
  long long Npad = (N + 255) & ~255LL;
  float* deg  = ws;
  float* h1   = deg  + Npad;
  float* agg1 = h1   + Npad * HID;
  float* h2   = agg1 + Npad * HID;

  const int TB = 256;
  auto blocks = [](long long work, int tb) { return (int)((work + tb - 1) / tb); };

  zero_f32<<<blocks(N, TB), TB, 0, stream>>>(deg, N);
  zero_f32<<<blocks(N * HID, TB), TB, 0, stream>>>(agg1, N * HID);

  degree_count<<<blocks(E, TB), TB, 0, stream>>>(col, deg, E);
  deg_to_dinv<<<blocks(N, TB), TB, 0, stream>>>(deg, (int)N);

  int mtiles = (int)((N + 15) / 16);
  int waves_per_block = TB / 32;
  int gblocks = (mtiles + waves_per_block - 1) / waves_per_block;

  gemm1_wmma<<<gblocks, TB, 0, stream>>>(x, W1, h1, mtiles, N);

  scatter_h1<<<blocks((long long)E * (HID / 4), TB), TB, 0, stream>>>(row, col, deg, h1, agg1, E);

  gemm2_wmma<<<gblocks, TB, 0, stream>>>(agg1, b1, W2, h2, mtiles, N);

  init_out_bias<<<blocks(N * OUT_C, TB), TB, 0, stream>>>(out, b2, N * OUT_C);

  scatter_h2<<<blocks((long long)E * (OUT_C / 4), TB), TB, 0, stream>>>(row, col, deg, h2, out, E);
}